// BaselineGDPModel_30812095381804
// MI455X (gfx1250) — compile-verified
//
#include <hip/hip_runtime.h>
#include <hip/hip_bf16.h>

// ---------------------------------------------------------------------------
// 2-layer GAT + linear head for MI455X (gfx1250, wave32).
// Dense transforms via V_WMMA_F32_16X16X4_F32 on padded node buffers
// (branchless); edge softmax/aggregation via fp32 global atomics
// (L2-resident working set: h/agg = 12.8MB each << 192MB L2).
// ---------------------------------------------------------------------------

typedef __attribute__((ext_vector_type(2))) float v2f;
typedef __attribute__((ext_vector_type(8))) float v8f;

#define HID 32
#define NEG_SLOPE 0.2f

// order-preserving float <-> uint mapping for atomic max
__device__ __forceinline__ unsigned f2ord(float f) {
    unsigned u = __float_as_uint(f);
    return (u & 0x80000000u) ? ~u : (u | 0x80000000u);
}
__device__ __forceinline__ float ord2f(unsigned u) {
    unsigned v = (u & 0x80000000u) ? (u & 0x7FFFFFFFu) : ~u;
    return __uint_as_float(v);
}
__device__ __forceinline__ float lrelu(float x) {
    return x > 0.0f ? x : NEG_SLOPE * x;
}

// ---------------------------------------------------------------------------
// Layer-1 transform: h[Npad,32] = x[N,3] @ W[3,32], K padded 3->4.
// One wave handles 16 nodes; 2 WMMA tiles cover the 32 output channels.
// h is padded to a multiple of 128 rows -> unconditional stores.
// A/B loads: clamped flat index + cndmask-zero (branchless).
// ---------------------------------------------------------------------------
__global__ void gat_transform1(const float* __restrict__ x,
                               const float* __restrict__ W,
                               float* __restrict__ h, int N) {
    int wave = (int)((blockIdx.x * blockDim.x + threadIdx.x) >> 5);
    int lane = threadIdx.x & 31;
    int half = lane >> 4;     // 0: K={0,1}, 1: K={2,3}
    int mrow = lane & 15;
    long base = (long)wave * 16;

    long nidx = base + mrow;
    if (nidx > N - 1) nidx = N - 1;        // v_cndmask clamp, no branch
    int ka = 2 * half;

    // unconditional in-bounds loads, then select-zero for the padded K=3 slot
    long xlim = (long)N * 3 - 1;
    long i0 = nidx * 3 + ka;     if (i0 > xlim) i0 = xlim;
    long i1 = nidx * 3 + ka + 1; if (i1 > xlim) i1 = xlim;
    float xa0 = x[i0];
    float xa1 = x[i1];
    v2f a;
    a.x = (ka + 0 < 3) ? xa0 : 0.0f;       // uniform per half: scalar select
    a.y = (ka + 1 < 3) ? xa1 : 0.0f;

#pragma unroll
    for (int t = 0; t < 2; ++t) {
        int col = t * 16 + mrow;
        int w0 = (ka + 0) * HID + col; if (w0 > 95) w0 = 95;
        int w1 = (ka + 1) * HID + col; if (w1 > 95) w1 = 95;
        float wb0 = W[w0];
        float wb1 = W[w1];
        v2f b;
        b.x = (ka + 0 < 3) ? wb0 : 0.0f;
        b.y = (ka + 1 < 3) ? wb1 : 0.0f;
        v8f c = {};
        c = __builtin_amdgcn_wmma_f32_16x16x4_f32(false, a, false, b,
                                                  (short)0, c, false, false);
#pragma unroll
        for (int r = 0; r < 8; ++r) {
            long node = base + r + 8 * half;   // always < Npad: store direct
            h[node * HID + col] = c[r];
        }
    }
}

// ---------------------------------------------------------------------------
// Layer-2 transform: h[Npad,32] = xin[Npad,32] @ W[32,32].
// One wave: 16 nodes x 32 cols, 8 K-steps of 16x16x4 per tile -> 16 WMMAs.
// Fully branchless (padded in/out buffers).
// ---------------------------------------------------------------------------
__global__ void gat_transform2(const float* __restrict__ xin,
                               const float* __restrict__ W,
                               float* __restrict__ h) {
    int wave = (int)((blockIdx.x * blockDim.x + threadIdx.x) >> 5);
    int lane = threadIdx.x & 31;
    int half = lane >> 4;
    int mrow = lane & 15;
    long base = (long)wave * 16;
    long nidx = base + mrow;                 // < Npad by construction

    v8f c0 = {}, c1 = {};
#pragma unroll
    for (int k0 = 0; k0 < HID; k0 += 4) {
        int ka = k0 + 2 * half;
        v2f a;
        a.x = xin[nidx * HID + ka + 0];      // contiguous pair -> b64 load
        a.y = xin[nidx * HID + ka + 1];
        v2f b0, b1;
        b0.x = W[(ka + 0) * HID + mrow];
        b0.y = W[(ka + 1) * HID + mrow];
        b1.x = W[(ka + 0) * HID + 16 + mrow];
        b1.y = W[(ka + 1) * HID + 16 + mrow];
        c0 = __builtin_amdgcn_wmma_f32_16x16x4_f32(false, a, false, b0,
                                                   (short)0, c0, false, false);
        c1 = __builtin_amdgcn_wmma_f32_16x16x4_f32(false, a, false, b1,
                                                   (short)0, c1, false, false);
    }
#pragma unroll
    for (int r = 0; r < 8; ++r) {
        long node = base + r + 8 * half;
        h[node * HID + mrow]      = c0[r];
        h[node * HID + 16 + mrow] = c1[r];
    }
}

// ---------------------------------------------------------------------------
// Per-node attention scores: a_s[n] = h[n,:] . att_s ; a_d[n] = h[n,:] . att_d
// One wave32 per node, lane = channel, butterfly reduce.
// ---------------------------------------------------------------------------
__global__ void gat_attn(const float* __restrict__ h,
                         const float* __restrict__ att_s,
                         const float* __restrict__ att_d,
                         float* __restrict__ a_s, float* __restrict__ a_d,
                         int N) {
    long wave = (blockIdx.x * (long)blockDim.x + threadIdx.x) >> 5;
    int lane = threadIdx.x & 31;
    if (wave >= N) return;
    float v = h[wave * HID + lane];
    float s = v * att_s[lane];
    float d = v * att_d[lane];
#pragma unroll
    for (int m = 16; m >= 1; m >>= 1) {
        s += __shfl_xor(s, m, 32);
        d += __shfl_xor(d, m, 32);
    }
    if (lane == 0) { a_s[wave] = s; a_d[wave] = d; }
}

// ---------------------------------------------------------------------------
// Zero-init mx (ordered-bits sentinel), denom, and the aggregation buffer.
// Grid covers N*32 elements.
// ---------------------------------------------------------------------------
__global__ void gat_init(unsigned* __restrict__ mx, float* __restrict__ denom,
                         float* __restrict__ agg, int N) {
    long i = blockIdx.x * (long)blockDim.x + threadIdx.x;
    long tot = (long)N * HID;
    if (i < tot) agg[i] = 0.0f;
    if (i < N) { mx[i] = 0u; denom[i] = 0.0f; }
}

// ---------------------------------------------------------------------------
// Edge pass 1: segment max of leaky_relu(a_s[src] + a_d[dst]) over dst.
// Edges [0,E) from edge_index, [E,E+N) are self-loops.
// ---------------------------------------------------------------------------
__global__ void gat_edge_max(const int* __restrict__ ei,
                             const float* __restrict__ a_s,
                             const float* __restrict__ a_d,
                             unsigned* __restrict__ mx, int E, int N) {
    long e = blockIdx.x * (long)blockDim.x + threadIdx.x;
    long T = (long)E + N;
    if (e >= T) return;
    int s, d;
    if (e < E) { s = ei[e]; d = ei[(long)E + e]; }
    else       { s = d = (int)(e - E); }
    float lg = lrelu(a_s[s] + a_d[d]);
    atomicMax(&mx[d], f2ord(lg));
}

// ---------------------------------------------------------------------------
// Edge pass 2: denom[dst] += exp(logit - m[dst])
// ---------------------------------------------------------------------------
__global__ void gat_edge_denom(const int* __restrict__ ei,
                               const float* __restrict__ a_s,
                               const float* __restrict__ a_d,
                               const unsigned* __restrict__ mx,
                               float* __restrict__ denom, int E, int N) {
    long e = blockIdx.x * (long)blockDim.x + threadIdx.x;
    long T = (long)E + N;
    if (e >= T) return;
    int s, d;
    if (e < E) { s = ei[e]; d = ei[(long)E + e]; }
    else       { s = d = (int)(e - E); }
    float lg = lrelu(a_s[s] + a_d[d]);
    float ex = __expf(lg - ord2f(mx[d]));
    atomicAdd(&denom[d], ex);
}

// ---------------------------------------------------------------------------
// Edge pass 3: agg[dst,:] += coef * h[src,:].  One wave per edge, lane=channel:
// 128B coalesced gather + 128B coalesced fp32 atomic-add segment per edge.
// ---------------------------------------------------------------------------
__global__ void gat_aggregate(const int* __restrict__ ei,
                              const float* __restrict__ a_s,
                              const float* __restrict__ a_d,
                              const unsigned* __restrict__ mx,
                              const float* __restrict__ denom,
                              const float* __restrict__ h,
                              float* __restrict__ agg, int E, int N) {
    long tid  = blockIdx.x * (long)blockDim.x + threadIdx.x;
    long wave = tid >> 5;
    int lane  = (int)(tid & 31);
    long T = (long)E + N;
    if (wave >= T) return;
    int s, d;
    if (wave < E) { s = ei[wave]; d = ei[(long)E + wave]; }
    else          { s = d = (int)(wave - E); }
    float lg   = lrelu(a_s[s] + a_d[d]);
    float ex   = __expf(lg - ord2f(mx[d]));
    float coef = ex / (denom[d] + 1e-16f);
    atomicAdd(&agg[(long)d * HID + lane], coef * h[(long)s * HID + lane]);
}

// ---------------------------------------------------------------------------
// Elementwise bias + ReLU, in place on the aggregation buffer.
// ---------------------------------------------------------------------------
__global__ void gat_bias_relu(float* __restrict__ agg,
                              const float* __restrict__ b, long total) {
    long i = blockIdx.x * (long)blockDim.x + threadIdx.x;
    if (i >= total) return;
    float v = agg[i] + b[i & (HID - 1)];
    agg[i] = v > 0.0f ? v : 0.0f;
}

// ---------------------------------------------------------------------------
// Head: out[n] = h[n,:] . Wl + bl. One wave per node.
// ---------------------------------------------------------------------------
__global__ void gat_final(const float* __restrict__ h,
                          const float* __restrict__ Wl,
                          const float* __restrict__ bl,
                          float* __restrict__ out, int N) {
    long wave = (blockIdx.x * (long)blockDim.x + threadIdx.x) >> 5;
    int lane = threadIdx.x & 31;
    if (wave >= N) return;
    float p = h[wave * HID + lane] * Wl[lane];
#pragma unroll
    for (int m = 16; m >= 1; m >>= 1) p += __shfl_xor(p, m, 32);
    if (lane == 0) out[wave] = p + bl[0];
}

// ---------------------------------------------------------------------------

static inline unsigned cdiv(long a, long b) { return (unsigned)((a + b - 1) / b); }

extern "C" void kernel_launch(void* const* d_in, const int* in_sizes, int n_in,
                              void* d_out, int out_size, void* d_ws, size_t ws_size,
                              hipStream_t stream) {
    const float* x        = (const float*)d_in[0];
    const int*   ei       = (const int*)  d_in[1];
    const float* W1       = (const float*)d_in[2];
    const float* att_src1 = (const float*)d_in[3];
    const float* att_dst1 = (const float*)d_in[4];
    const float* b1       = (const float*)d_in[5];
    const float* W2       = (const float*)d_in[6];
    const float* att_src2 = (const float*)d_in[7];
    const float* att_dst2 = (const float*)d_in[8];
    const float* b2       = (const float*)d_in[9];
    const float* Wl       = (const float*)d_in[10];
    const float* bl       = (const float*)d_in[11];
    float* out = (float*)d_out;

    const int N = in_sizes[0] / 3;
    const int E = in_sizes[1] / 2;
    const long Npad = ((long)N + 127) / 128 * 128;   // WMMA-block padded rows

    // workspace layout (floats); node matrices padded to Npad rows
    float*    bufA  = (float*)d_ws;               // h (transform out / messages)
    float*    bufB  = bufA + Npad * HID;          // x2 / agg (ping-pong)
    float*    a_s   = bufB + Npad * HID;
    float*    a_d   = a_s + N;
    unsigned* mx    = (unsigned*)(a_d + N);
    float*    denom = (float*)(mx + N);

    const int BLK = 256;
    const long TOTAL_E = (long)E + N;            // edges incl. self-loops
    const unsigned gNodesWmma = (unsigned)(Npad / 128); // 8 waves * 16 nodes
    const unsigned gNodesWave = cdiv(N, 8);      // wave per node
    const unsigned gElems     = cdiv((long)N * HID, BLK);
    const unsigned gEdges     = cdiv(TOTAL_E, BLK);
    const unsigned gEdgeWaves = cdiv(TOTAL_E * 32, BLK);

    // ---------------- layer 1 ----------------
    gat_transform1<<<gNodesWmma, BLK, 0, stream>>>(x, W1, bufA, N);
    gat_attn<<<gNodesWave, BLK, 0, stream>>>(bufA, att_src1, att_dst1, a_s, a_d, N);
    gat_init<<<gElems, BLK, 0, stream>>>(mx, denom, bufB, N);
    gat_edge_max<<<gEdges, BLK, 0, stream>>>(ei, a_s, a_d, mx, E, N);
    gat_edge_denom<<<gEdges, BLK, 0, stream>>>(ei, a_s, a_d, mx, denom, E, N);
    gat_aggregate<<<gEdgeWaves, BLK, 0, stream>>>(ei, a_s, a_d, mx, denom, bufA, bufB, E, N);
    gat_bias_relu<<<gElems, BLK, 0, stream>>>(bufB, b1, (long)N * HID);

    // ---------------- layer 2 ----------------
    // padded tail rows of bufB hold stale data; transform2 reads them but the
    // corresponding h rows are never consumed downstream.
    gat_transform2<<<gNodesWmma, BLK, 0, stream>>>(bufB, W2, bufA);
    gat_attn<<<gNodesWave, BLK, 0, stream>>>(bufA, att_src2, att_dst2, a_s, a_d, N);
    gat_init<<<gElems, BLK, 0, stream>>>(mx, denom, bufB, N);  // bufB free now
    gat_edge_max<<<gEdges, BLK, 0, stream>>>(ei, a_s, a_d, mx, E, N);
    gat_edge_denom<<<gEdges, BLK, 0, stream>>>(ei, a_s, a_d, mx, denom, E, N);
    gat_aggregate<<<gEdgeWaves, BLK, 0, stream>>>(ei, a_s, a_d, mx, denom, bufA, bufB, E, N);
    gat_bias_relu<<<gElems, BLK, 0, stream>>>(bufB, b2, (long)N * HID);

    // ---------------- head ----------------
    gat_final<<<gNodesWave, BLK, 0, stream>>>(bufB, Wl, bl, out, N);
}